// MHAttentionModel_455266534014
// MI455X (gfx1250) — compile-verified
//
#include <hip/hip_runtime.h>
#include <hip/hip_bf16.h>
#include <math.h>

typedef __attribute__((ext_vector_type(2))) float v2f;
typedef __attribute__((ext_vector_type(8))) float v8f;

#define WSZ 512      // window length
#define NB  16       // batch
#define NH  16       // heads
#define HD  16       // head dim
#define EPSBN 1e-5f

__device__ __forceinline__ v8f wmma4(v2f a, v2f b, v8f c) {
  // V_WMMA_F32_16X16X4_F32 : D = A(16x4 f32) x B(4x16 f32) + C(16x16 f32)
  return __builtin_amdgcn_wmma_f32_16x16x4_f32(false, a, false, b, (short)0, c,
                                               false, false);
}

// ---------------------------------------------------------------------------
// Conv1d(same pad) + eval BatchNorm + ReLU as a WMMA GEMM with 2x2 register
// blocking per wave: 32 cout x 32 positions per wave. Per K-step: 2 b64 weight
// loads + 4 b32 im2col loads feed 4 WMMAs (vs 12 loads / 4 WMMA unblocked).
// A = W (Cout x Cin*KS row-major), B = im2col patch (Cin*KS x positions).
// ---------------------------------------------------------------------------
template <int KS>
__global__ void conv_bn_relu_wmma(const float* __restrict__ X,   // (N,Cin,W)
                                  const float* __restrict__ Wt,  // (Cout,Cin,KS)
                                  const float* __restrict__ bias,
                                  const float* __restrict__ gamma,
                                  const float* __restrict__ beta,
                                  const float* __restrict__ mean,
                                  const float* __restrict__ var,
                                  float* __restrict__ Y,         // (N,Cout,W)
                                  int Cin, int Cout) {
  const int lane = threadIdx.x & 31;
  const int mh   = lane >> 4;    // lane-half selects K offset pair
  const int lN   = lane & 15;
  const int coT  = blockIdx.x * 32;   // 2 cout tiles
  const int tT   = blockIdx.y * 32;   // 2 position tiles
  const int n    = blockIdx.z;
  const int pad  = (KS - 1) / 2;
  const int K    = Cin * KS;          // multiple of 4 for all layers here

  const float* Xn = X + (size_t)n * Cin * WSZ;
  const int t0 = tT + lN;
  const int t1 = tT + 16 + lN;

  const float* wrow0 = Wt + (size_t)(coT + lN) * K;
  const float* wrow1 = Wt + (size_t)(coT + 16 + lN) * K;

  v8f acc00 = {}, acc01 = {}, acc10 = {}, acc11 = {};
  for (int k = 0; k < K; k += 4) {
    const int koff = k + 2 * mh;   // even -> 8B-aligned within each row
    // --- A fragments: weight rows are contiguous -> single b64 loads ---
    const v2f a0 = *(const v2f*)(wrow0 + koff);
    const v2f a1 = *(const v2f*)(wrow1 + koff);
    __builtin_prefetch(wrow0 + koff + 64, 0, 1);  // global_prefetch_b8 stream
    // --- B fragments: im2col rows {koff, koff+1}, cols t0 / t1 ---
    const int kk0 = koff, kk1 = koff + 1;
    const int ci0 = kk0 / KS, j0 = kk0 - ci0 * KS;
    const int ci1 = kk1 / KS, j1 = kk1 - ci1 * KS;
    v2f b0, b1;
    {
      const int s00 = t0 - pad + j0, s01 = t0 - pad + j1;
      b0.x = ((unsigned)s00 < (unsigned)WSZ) ? Xn[(size_t)ci0 * WSZ + s00] : 0.0f;
      b0.y = ((unsigned)s01 < (unsigned)WSZ) ? Xn[(size_t)ci1 * WSZ + s01] : 0.0f;
      const int s10 = t1 - pad + j0, s11 = t1 - pad + j1;
      b1.x = ((unsigned)s10 < (unsigned)WSZ) ? Xn[(size_t)ci0 * WSZ + s10] : 0.0f;
      b1.y = ((unsigned)s11 < (unsigned)WSZ) ? Xn[(size_t)ci1 * WSZ + s11] : 0.0f;
    }
    acc00 = wmma4(a0, b0, acc00);
    acc01 = wmma4(a0, b1, acc01);
    acc10 = wmma4(a1, b0, acc10);
    acc11 = wmma4(a1, b1, acc11);
  }

  // Epilogue: C layout -> lane holds co = tile + r + 8*mh, t = tile + lN
#pragma unroll
  for (int ct = 0; ct < 2; ++ct) {
#pragma unroll
    for (int r = 0; r < 8; ++r) {
      const int co = coT + 16 * ct + r + 8 * mh;
      const float sc = gamma[co] * rsqrtf(var[co] + EPSBN);
      const float sh = (bias[co] - mean[co]) * sc + beta[co];
      const float c0 = (ct == 0) ? acc00[r] : acc10[r];
      const float c1 = (ct == 0) ? acc01[r] : acc11[r];
      float* yrow = Y + ((size_t)n * Cout + co) * WSZ;
      yrow[t0] = fmaxf(c0 * sc + sh, 0.0f);
      yrow[t1] = fmaxf(c1 * sc + sh, 0.0f);
    }
  }
}

// ---------------------------------------------------------------------------
// Attention pass 1: per (n,h) and 16-column tile, stream S = Xh^T Xh over the
// l dimension via WMMA and keep an online (max,sum) of exp(relu(s*S+bias)).
// Column stats indexed by the summed dim b of the output einsum.
// ---------------------------------------------------------------------------
__global__ void attn_stats_wmma(const float* __restrict__ X,  // (N, NH*HD, W)
                                const float* __restrict__ biasScalar,
                                float* __restrict__ Mbuf,     // (N,NH,W)
                                float* __restrict__ Dbuf) {
  const int lane = threadIdx.x & 31;
  const int mh   = lane >> 4;
  const int lN   = lane & 15;
  const int bT   = blockIdx.x * 16;
  const int h    = blockIdx.y;
  const int n    = blockIdx.z;
  const float scaleS = 0.25f;               // 1/sqrt(HD)
  const float bv = biasScalar[0];

  const float* Xh = X + ((size_t)n * (NH * HD) + h * HD) * WSZ;  // (HD x W)

  float m_run = -1e30f, d_run = 0.0f;
  for (int lT = 0; lT < WSZ; lT += 16) {
    v8f c = {};
#pragma unroll
    for (int s = 0; s < 4; ++s) {
      const int p0 = 4 * s + 2 * mh;
      v2f a, b;
      // A[m=l, k=p] = Xh[p, lT+m]
      a.x = Xh[(size_t)p0 * WSZ + lT + lN];
      a.y = Xh[(size_t)(p0 + 1) * WSZ + lT + lN];
      // B[k=p, n=b] = Xh[p, bT+n]
      b.x = Xh[(size_t)p0 * WSZ + bT + lN];
      b.y = Xh[(size_t)(p0 + 1) * WSZ + bT + lN];
      c = wmma4(a, b, c);
    }
    // c[r] = S[l = lT + r + 8*mh, b = bT + lN]
    float e[8];
    float tmax = m_run;
#pragma unroll
    for (int r = 0; r < 8; ++r) {
      e[r] = fmaxf(scaleS * c[r] + bv, 0.0f);
      tmax = fmaxf(tmax, e[r]);
    }
    float d_new = d_run * __expf(m_run - tmax);
#pragma unroll
    for (int r = 0; r < 8; ++r) d_new += __expf(e[r] - tmax);
    m_run = tmax;
    d_run = d_new;
  }
  // lanes L and L^16 hold the same column b -> combine
  const float m_o = __shfl_xor(m_run, 16, 32);
  const float d_o = __shfl_xor(d_run, 16, 32);
  const float m_f = fmaxf(m_run, m_o);
  const float d_f = d_run * __expf(m_run - m_f) + d_o * __expf(m_o - m_f);
  if (mh == 0) {
    const size_t idx = ((size_t)n * NH + h) * WSZ + bT + lN;
    Mbuf[idx] = m_f;
    Dbuf[idx] = d_f;
  }
}

// ---------------------------------------------------------------------------
// Attention pass 2: Xt[c,a] = sum_b  exp(relu(s*S[a,b]+bias)-M[b])/D[b] * Xh[c,b]
// S tile recomputed via WMMA (as S[b,a] so stats index is on the VGPR-row axis),
// converted to P, transposed through LDS into B-operand layout, second WMMA
// chain accumulates Xh * P^T. Writes channels [256+h*16 .. ) of the concat.
// ---------------------------------------------------------------------------
__global__ void attn_out_wmma(const float* __restrict__ X,  // (N, NH*HD, W)
                              const float* __restrict__ biasScalar,
                              const float* __restrict__ Mbuf,
                              const float* __restrict__ Dbuf,
                              float* __restrict__ Out) {     // (N, 2*NH*HD, W)
  __shared__ float Plds[16][17];  // [b_local][a_local], padded
  const int lane = threadIdx.x & 31;
  const int mh   = lane >> 4;
  const int lN   = lane & 15;
  const int aT   = blockIdx.x * 16;
  const int h    = blockIdx.y;
  const int n    = blockIdx.z;
  const float scaleS = 0.25f;
  const float bv = biasScalar[0];

  const float* Xh   = X + ((size_t)n * (NH * HD) + h * HD) * WSZ;
  const float* Mrow = Mbuf + ((size_t)n * NH + h) * WSZ;
  const float* Drow = Dbuf + ((size_t)n * NH + h) * WSZ;

  v8f acc = {};
  for (int bT = 0; bT < WSZ; bT += 16) {
    // ---- S tile, rows = b, cols = a ----
    v8f c = {};
#pragma unroll
    for (int s = 0; s < 4; ++s) {
      const int p0 = 4 * s + 2 * mh;
      v2f a, b;
      a.x = Xh[(size_t)p0 * WSZ + bT + lN];        // A[m=b,k=p]
      a.y = Xh[(size_t)(p0 + 1) * WSZ + bT + lN];
      b.x = Xh[(size_t)p0 * WSZ + aT + lN];        // B[k=p,n=a]
      b.y = Xh[(size_t)(p0 + 1) * WSZ + aT + lN];
      c = wmma4(a, b, c);
    }
    // c[r] = S[b = bT + r + 8*mh, a = aT + lN] ; P = exp(relu(.)-M[b])/D[b]
#pragma unroll
    for (int r = 0; r < 8; ++r) {
      const int b = bT + r + 8 * mh;
      const float e = fmaxf(scaleS * c[r] + bv, 0.0f);
      Plds[r + 8 * mh][lN] = __expf(e - Mrow[b]) / Drow[b];
    }
    asm volatile("s_wait_dscnt 0" ::: "memory");
    // ---- accumulate Xt[c,a] += Xh[c,b] * P[b,a] over this b tile ----
#pragma unroll
    for (int s = 0; s < 4; ++s) {
      const int kb = 4 * s + 2 * mh;               // even -> 8B-aligned
      v2f a2, b2;
      a2 = *(const v2f*)(Xh + (size_t)lN * WSZ + bT + kb);  // A[m=c,k=b], b64
      b2.x = Plds[kb][lN];                          // B[k=b,n=a]
      b2.y = Plds[kb + 1][lN];
      acc = wmma4(a2, b2, acc);
    }
  }
  // acc[r] = Xt[c = r + 8*mh, a = aT + lN]; concat second half of channels
#pragma unroll
  for (int r = 0; r < 8; ++r) {
    const int cc = r + 8 * mh;
    Out[((size_t)n * (2 * NH * HD) + (NH * HD) + h * HD + cc) * WSZ + aT + lN] =
        acc[r];
  }
}

// Copy X (N,256,W) into concat buffer channels [0,256)
__global__ void copy_into_cat(const float* __restrict__ X,
                              float* __restrict__ Out) {
  const size_t i = (size_t)blockIdx.x * blockDim.x + threadIdx.x;
  const size_t total = (size_t)NB * 256 * WSZ;
  if (i < total) {
    const int t = (int)(i & (WSZ - 1));
    const size_t rest = i >> 9;
    const int ch = (int)(rest & 255);
    const int n = (int)(rest >> 8);
    Out[((size_t)n * 512 + ch) * WSZ + t] = X[i];
  }
}

// ---------------------------------------------------------------------------
extern "C" void kernel_launch(void* const* d_in, const int* in_sizes, int n_in,
                              void* d_out, int out_size, void* d_ws,
                              size_t ws_size, hipStream_t stream) {
  (void)in_sizes; (void)n_in; (void)out_size; (void)ws_size;

  const float* x = (const float*)d_in[0];
  // layer i params at d_in[1 + 6*i + {0..5}] = w,b,gamma,beta,mean,var
  auto W  = [&](int i) { return (const float*)d_in[1 + 6 * i + 0]; };
  auto Bs = [&](int i) { return (const float*)d_in[1 + 6 * i + 1]; };
  auto G  = [&](int i) { return (const float*)d_in[1 + 6 * i + 2]; };
  auto Be = [&](int i) { return (const float*)d_in[1 + 6 * i + 3]; };
  auto Mn = [&](int i) { return (const float*)d_in[1 + 6 * i + 4]; };
  auto Vr = [&](int i) { return (const float*)d_in[1 + 6 * i + 5]; };
  const float* ab0 = (const float*)d_in[31];
  const float* ab1 = (const float*)d_in[32];

  char* ws = (char*)d_ws;
  const size_t szBuf = (size_t)NB * 256 * WSZ * sizeof(float);  // 8 MB
  const size_t szCat = (size_t)NB * 512 * WSZ * sizeof(float);  // 16 MB
  const size_t szStat = (size_t)NB * NH * WSZ * sizeof(float);  // 0.5 MB
  float* bufA = (float*)(ws);
  float* bufB = (float*)(ws + szBuf);
  float* cat  = (float*)(ws + 2 * szBuf);
  float* Mb   = (float*)(ws + 2 * szBuf + szCat);
  float* Db   = (float*)(ws + 2 * szBuf + szCat + szStat);
  float* out  = (float*)d_out;

  const dim3 blk(32);
  const dim3 gC256(256 / 32, WSZ / 32, NB);   // 2x2 blocked conv tiles
  const dim3 gC128(128 / 32, WSZ / 32, NB);
  const dim3 gAtt(WSZ / 16, NH, NB);
  const dim3 gCopy((unsigned)(((size_t)NB * 256 * WSZ + 255) / 256));

  // conv0: 128->256, k=1
  conv_bn_relu_wmma<1><<<gC256, blk, 0, stream>>>(
      x, W(0), Bs(0), G(0), Be(0), Mn(0), Vr(0), bufA, 128, 256);
  // conv1: 256->256, k=1
  conv_bn_relu_wmma<1><<<gC256, blk, 0, stream>>>(
      bufA, W(1), Bs(1), G(1), Be(1), Mn(1), Vr(1), bufB, 256, 256);
  // conv2: 256->256, k=3
  conv_bn_relu_wmma<3><<<gC256, blk, 0, stream>>>(
      bufB, W(2), Bs(2), G(2), Be(2), Mn(2), Vr(2), bufA, 256, 256);

  // attention 1: bufA -> cat (16,512,512)
  attn_stats_wmma<<<gAtt, blk, 0, stream>>>(bufA, ab0, Mb, Db);
  copy_into_cat<<<gCopy, dim3(256), 0, stream>>>(bufA, cat);
  attn_out_wmma<<<gAtt, blk, 0, stream>>>(bufA, ab0, Mb, Db, cat);

  // conv3: 512->256, k=9
  conv_bn_relu_wmma<9><<<gC256, blk, 0, stream>>>(
      cat, W(3), Bs(3), G(3), Be(3), Mn(3), Vr(3), bufB, 512, 256);

  // attention 2: bufB -> cat
  attn_stats_wmma<<<gAtt, blk, 0, stream>>>(bufB, ab1, Mb, Db);
  copy_into_cat<<<gCopy, dim3(256), 0, stream>>>(bufB, cat);
  attn_out_wmma<<<gAtt, blk, 0, stream>>>(bufB, ab1, Mb, Db, cat);

  // conv4: 512->128, k=19 -> output
  conv_bn_relu_wmma<19><<<gC128, blk, 0, stream>>>(
      cat, W(4), Bs(4), G(4), Be(4), Mn(4), Vr(4), out, 512, 128);
}